// GAT_3135326126437
// MI455X (gfx1250) — compile-verified
//
#include <hip/hip_runtime.h>
#include <hip/hip_bf16.h>
#include <math.h>

// ---------------------------------------------------------------------------
// GAT (2-layer, 8 heads, D=64) for gfx1250.
//  - Dense GEMMs: native fp32 WMMA (v_wmma_f32_16x16x4_f32), double-buffered
//    LDS tiles staged with CDNA5 async copies (global_load_async_to_lds_b128,
//    s_wait_asynccnt) so HBM traffic overlaps the matrix pipes.
//  - Edge softmax / SpMM exploit sorted edge_src via CSR (no atomics).
// ---------------------------------------------------------------------------

#define ALPHA 0.2f
#define HEADS 8
#define DHEAD 64
#define HD    (HEADS * DHEAD)   // 512

typedef __attribute__((ext_vector_type(2))) float v2f;
typedef __attribute__((ext_vector_type(8))) float v8f;

#define BM 128
#define BN 64
#define KT 16
#define AP 20   // A row pad: 80B rows -> 16B aligned, conflict-free column reads
#define BP 72   // B row pad: 288B rows -> 16B aligned, half-wave banks disjoint

// Low 32 bits of an LDS pointer = byte offset within LDS aperture.
__device__ __forceinline__ unsigned ldsOff(const void* p) {
    return (unsigned)(unsigned long long)p;
}
// CDNA5 async DMA: 16B global -> 16B LDS, tracked by ASYNCcnt.
__device__ __forceinline__ void asyncLoadB128(unsigned loff, const float* g) {
    asm volatile("global_load_async_to_lds_b128 %0, %1, off"
                 :: "v"(loff), "v"(g) : "memory");
}
__device__ __forceinline__ void waitAsyncLE3() {
    asm volatile("s_wait_asynccnt 0x3" ::: "memory");
}
__device__ __forceinline__ void waitAsync0() {
    asm volatile("s_wait_asynccnt 0x0" ::: "memory");
}

// C = A[M,K] @ B + bias, where B is either
//   mode 0: plain row-major [K, Ncols]
//   mode 1: head-blocked weight [H][K][64], column n -> (n>>6)*K*64 + k*64 + (n&63)
// Requires: K % 16 == 0, Ncols % 64 == 0 (and >= col0+64), 16B-aligned A/B.
__global__ __launch_bounds__(256) void gemm_f32_wmma(
    const float* __restrict__ A, const float* __restrict__ B,
    const float* __restrict__ bias, float* __restrict__ C,
    int M, int K, int Ncols, int mode)
{
    __shared__ __align__(16) float As[2][BM][AP];
    __shared__ __align__(16) float Bs[2][KT][BP];

    const int tid  = threadIdx.x;
    const int lane = tid & 31;          // wave32
    const int wave = tid >> 5;          // 0..7
    const int row0 = blockIdx.y * BM;
    const int col0 = blockIdx.x * BN;

    v8f c[4];
    #pragma unroll
    for (int j = 0; j < 4; ++j)
        #pragma unroll
        for (int i = 0; i < 8; ++i) c[j][i] = 0.0f;

    // Issue one K-tile's staging: uniform 3 async b128 per thread
    // (2 A chunks + 1 B chunk), so ASYNCcnt thresholds are exact.
    auto issueTile = [&](int k0, int buf) {
        #pragma unroll
        for (int s = 0; s < 2; ++s) {
            const int q  = tid + s * 256;    // chunk id 0..511
            const int r  = q >> 2;           // tile row 0..127
            const int ck = q & 3;            // 4-float chunk within KT
            int grow = row0 + r;
            if (grow >= M) grow = M - 1;     // clamp: rows >= M never stored
            asyncLoadB128(ldsOff(&As[buf][r][ck * 4]),
                          A + (size_t)grow * K + (k0 + ck * 4));
        }
        const int gk = tid >> 4;             // tile k-row 0..15
        const int cn = tid & 15;             // 4-float chunk within BN
        const float* brow = (mode == 0)
            ? B + (size_t)(k0 + gk) * Ncols + col0
            : B + (size_t)(col0 >> 6) * ((size_t)K * 64) + (size_t)(k0 + gk) * 64;
        asyncLoadB128(ldsOff(&Bs[buf][gk][cn * 4]), brow + cn * 4);
    };

    issueTile(0, 0);
    const int ntiles = K / KT;

    for (int t = 0; t < ntiles; ++t) {
        const int cur = t & 1;
        if (t + 1 < ntiles) {
            issueTile((t + 1) * KT, cur ^ 1);   // prefetch next tile into other buffer
            waitAsyncLE3();                     // drain previous batch (<= just-issued 3)
        } else {
            waitAsync0();
        }
        __syncthreads();

        const int wr = wave * 16;
        #pragma unroll
        for (int kk = 0; kk < KT; kk += 4) {
            // A fragment (16x4): lanes 0-15 hold K={kk,kk+1}, lanes 16-31 K={kk+2,kk+3}
            const int mrow = wr + (lane & 15);
            const int ksel = kk + ((lane >> 4) << 1);
            v2f a;
            a.x = As[cur][mrow][ksel];          // contiguous pair -> ds_load_b64
            a.y = As[cur][mrow][ksel + 1];
            #pragma unroll
            for (int j = 0; j < 4; ++j) {
                const int ncol = j * 16 + (lane & 15);
                v2f b;
                b.x = Bs[cur][ksel][ncol];
                b.y = Bs[cur][ksel + 1][ncol];
                c[j] = __builtin_amdgcn_wmma_f32_16x16x4_f32(
                    false, a, false, b, (short)0, c[j], false, false);
            }
        }
        __syncthreads();   // all waves done with As/Bs[cur] before it is re-filled
    }

    // epilogue: C/D layout — VGPR r: lanes0-15 M=r, lanes16-31 M=r+8, N=lane&15
    const int wr = wave * 16;
    #pragma unroll
    for (int j = 0; j < 4; ++j) {
        #pragma unroll
        for (int r = 0; r < 8; ++r) {
            const int grow = row0 + wr + r + ((lane >> 4) << 3);
            const int gcol = col0 + j * 16 + (lane & 15);
            if (grow < M) {
                float v = c[j][r];
                if (bias) v += bias[gcol];
                C[(size_t)grow * Ncols + gcol] = v;
            }
        }
    }
}

// f1[n,h] = sum_d Wh[n,h,d]*a_s[h,d];  f2 likewise with a_d
__global__ void gat_scores(const float* __restrict__ Wh,
                           const float* __restrict__ a_s,
                           const float* __restrict__ a_d,
                           float* __restrict__ f1, float* __restrict__ f2, int N)
{
    const int i = blockIdx.x * blockDim.x + threadIdx.x;
    if (i >= N * HEADS) return;
    const int h = i & (HEADS - 1);
    const int n = i >> 3;
    const float* w  = Wh + (size_t)n * HD + h * DHEAD;
    const float* as = a_s + h * DHEAD;
    const float* ad = a_d + h * DHEAD;
    float s1 = 0.f, s2 = 0.f;
    #pragma unroll 8
    for (int d = 0; d < DHEAD; ++d) { const float v = w[d]; s1 += v * as[d]; s2 += v * ad[d]; }
    f1[i] = s1;
    f2[i] = s2;
}

// Build CSR row pointers from sorted edge_src.
__global__ void build_rowptr(const int* __restrict__ src, int* __restrict__ row_start,
                             int N, int E)
{
    const int e = blockIdx.x * blockDim.x + threadIdx.x;
    if (e >= E) return;
    const int s    = src[e];
    const int prev = (e == 0) ? -1 : src[e - 1];
    for (int n = prev + 1; n <= s; ++n) row_start[n] = e;
    if (e == E - 1)
        for (int n = s + 1; n <= N; ++n) row_start[n] = E;
}

// Per (node, head): segmented LeakyReLU + softmax over the node's edges.
__global__ void edge_softmax(const int* __restrict__ dst, const int* __restrict__ row_start,
                             const float* __restrict__ f1, const float* __restrict__ f2,
                             float* __restrict__ att, int N)
{
    const int i = blockIdx.x * blockDim.x + threadIdx.x;
    if (i >= N * HEADS) return;
    const int h  = i & (HEADS - 1);
    const int n  = i >> 3;
    const int e0 = row_start[n], e1 = row_start[n + 1];
    if (e0 >= e1) return;
    const float f1v = f1[i];

    float m = -INFINITY;
    for (int e = e0; e < e1; ++e) {
        float sc = f1v + f2[dst[e] * HEADS + h];
        sc = sc > 0.f ? sc : ALPHA * sc;
        m = fmaxf(m, sc);
    }
    float ssum = 0.f;
    for (int e = e0; e < e1; ++e) {
        float sc = f1v + f2[dst[e] * HEADS + h];
        sc = sc > 0.f ? sc : ALPHA * sc;
        const float p = expf(sc - m);
        att[(size_t)e * HEADS + h] = p;
        ssum += p;
    }
    const float inv = 1.0f / fmaxf(ssum, 1e-16f);
    for (int e = e0; e < e1; ++e) att[(size_t)e * HEADS + h] *= inv;
}

// One 512-thread block per node: out[n, t] = ELU( sum_e att[e, t/64] * Wh[dst[e], t] )
__global__ __launch_bounds__(512) void spmm_elu(
    const int* __restrict__ dst, const int* __restrict__ row_start,
    const float* __restrict__ att, const float* __restrict__ Wh,
    float* __restrict__ out, int N)
{
    const int n = blockIdx.x;
    const int t = threadIdx.x;          // 0..511 -> (h = t>>6, d = t&63)
    const int h = t >> 6;
    const int e0 = row_start[n], e1 = row_start[n + 1];
    float acc = 0.f;
    for (int e = e0; e < e1; ++e) {
        const float a = att[(size_t)e * HEADS + h];
        acc += a * Wh[(size_t)dst[e] * HD + t];   // coalesced 2KB row gather
    }
    acc = acc > 0.f ? acc : (expf(acc) - 1.0f);   // ELU
    out[(size_t)n * HD + t] = acc;
}

extern "C" void kernel_launch(void* const* d_in, const int* in_sizes, int n_in,
                              void* d_out, int out_size, void* d_ws, size_t ws_size,
                              hipStream_t stream)
{
    const float* x        = (const float*)d_in[0];
    const int*   edge_src = (const int*)  d_in[1];
    const int*   edge_dst = (const int*)  d_in[2];
    const float* W1       = (const float*)d_in[3];
    const float* a_src1   = (const float*)d_in[4];
    const float* a_dst1   = (const float*)d_in[5];
    const float* W2       = (const float*)d_in[6];
    const float* a_src2   = (const float*)d_in[7];
    const float* a_dst2   = (const float*)d_in[8];
    const float* lin_W    = (const float*)d_in[9];
    const float* lin_b    = (const float*)d_in[10];

    const int N = in_sizes[0] / 256;    // F_IN = 256
    const int E = in_sizes[1];
    const int F_IN = 256;

    // workspace layout
    float* Wh  = (float*)d_ws;                    // [N, 512] (reused both layers)
    float* h1  = Wh  + (size_t)N * HD;            // [N, 512]
    float* h2  = h1  + (size_t)N * HD;            // [N, 512]
    float* f1  = h2  + (size_t)N * HD;            // [N, 8]
    float* f2  = f1  + (size_t)N * HEADS;         // [N, 8]
    float* att = f2  + (size_t)N * HEADS;         // [E, 8]
    int*   row_start = (int*)(att + (size_t)E * HEADS);  // [N+1]

    const dim3 blk256(256), blk512(512);
    const int nhBlocks = (N * HEADS + 255) / 256;

    // CSR from sorted src (shared by both layers)
    build_rowptr<<<(E + 255) / 256, blk256, 0, stream>>>(edge_src, row_start, N, E);

    // ---- layer 1 ----
    {
        dim3 grid(HD / BN, (N + BM - 1) / BM);
        gemm_f32_wmma<<<grid, blk256, 0, stream>>>(x, W1, nullptr, Wh, N, F_IN, HD, 1);
    }
    gat_scores<<<nhBlocks, blk256, 0, stream>>>(Wh, a_src1, a_dst1, f1, f2, N);
    edge_softmax<<<nhBlocks, blk256, 0, stream>>>(edge_dst, row_start, f1, f2, att, N);
    spmm_elu<<<N, blk512, 0, stream>>>(edge_dst, row_start, att, Wh, h1, N);

    // ---- layer 2 ----
    {
        dim3 grid(HD / BN, (N + BM - 1) / BM);
        gemm_f32_wmma<<<grid, blk256, 0, stream>>>(h1, W2, nullptr, Wh, N, HD, HD, 1);
    }
    gat_scores<<<nhBlocks, blk256, 0, stream>>>(Wh, a_src2, a_dst2, f1, f2, N);
    edge_softmax<<<nhBlocks, blk256, 0, stream>>>(edge_dst, row_start, f1, f2, att, N);
    spmm_elu<<<N, blk512, 0, stream>>>(edge_dst, row_start, att, Wh, h2, N);

    // ---- output projection: [N,512] @ [512,64] + b ----
    {
        dim3 grid(64 / BN, (N + BM - 1) / BM);
        gemm_f32_wmma<<<grid, blk256, 0, stream>>>(h2, lin_W, lin_b, (float*)d_out,
                                                   N, HD, 64, 0);
    }
}